// SparsemaxN_73521250173289
// MI455X (gfx1250) — compile-verified
//
#include <hip/hip_runtime.h>
#include <hip/hip_bf16.h>
#include <cstdint>

#define TPB    512
#define NWAVES (TPB / 32)
#define MAXN   8192

typedef __attribute__((ext_vector_type(4))) unsigned int u32x4;
typedef __attribute__((ext_vector_type(4))) int          i32x4;
typedef __attribute__((ext_vector_type(8))) int          i32x8;

// ---------------- prefix scan of segment sizes -> exclusive offsets ----------
// One block of B/2 (=1024) threads; each thread handles 2 segments.
__global__ void seg_offsets_kernel(const long long* __restrict__ sizes,
                                   unsigned long long* __restrict__ offs) {
    __shared__ unsigned long long tmp[1024];
    const int t = threadIdx.x;
    unsigned long long a = (unsigned long long)sizes[2 * t];
    unsigned long long b = (unsigned long long)sizes[2 * t + 1];
    unsigned long long s = a + b;
    tmp[t] = s;
    __syncthreads();
    for (int d = 1; d < 1024; d <<= 1) {
        unsigned long long v = (t >= d) ? tmp[t - d] : 0ull;
        __syncthreads();
        tmp[t] += v;
        __syncthreads();
    }
    unsigned long long excl = tmp[t] - s;   // exclusive prefix for segment 2t
    offs[2 * t]     = excl;
    offs[2 * t + 1] = excl + a;
}

// ---------------- block reductions (wave32 shuffles + LDS combine) -----------
__device__ __forceinline__ float wave_sum(float v) {
    for (int o = 16; o > 0; o >>= 1) v += __shfl_down(v, o, 32);
    return v;
}
__device__ __forceinline__ float wave_max(float v) {
    for (int o = 16; o > 0; o >>= 1) v = fmaxf(v, __shfl_down(v, o, 32));
    return v;
}
__device__ __forceinline__ float block_sum(float v, float* red) {
    const int lane = threadIdx.x & 31, w = threadIdx.x >> 5;
    v = wave_sum(v);
    __syncthreads();
    if (lane == 0) red[w] = v;
    __syncthreads();
    if (w == 0) {
        float x = (lane < NWAVES) ? red[lane] : 0.0f;
        for (int o = NWAVES / 2; o > 0; o >>= 1) x += __shfl_down(x, o, 32);
        if (lane == 0) red[NWAVES] = x;
    }
    __syncthreads();
    return red[NWAVES];
}
__device__ __forceinline__ float block_max(float v, float* red) {
    const int lane = threadIdx.x & 31, w = threadIdx.x >> 5;
    v = wave_max(v);
    __syncthreads();
    if (lane == 0) red[w] = v;
    __syncthreads();
    if (w == 0) {
        float x = (lane < NWAVES) ? red[lane] : -3.402823466e38f;
        for (int o = NWAVES / 2; o > 0; o >>= 1) x = fmaxf(x, __shfl_down(x, o, 32));
        if (lane == 0) red[NWAVES] = x;
    }
    __syncthreads();
    return red[NWAVES];
}

// ---------------- TDM descriptor (ISA cdna5 §8): 1-row tile of n fp32 --------
#if __has_builtin(__builtin_amdgcn_tensor_load_to_lds)
#define HAVE_TDM 1
__device__ __forceinline__ void make_tdm_desc(u32x4& g0, i32x8& g1,
                                              const void* gptr,
                                              unsigned lds_base, int n) {
    unsigned long long ga = (unsigned long long)(uintptr_t)gptr;  // 57-bit VA
    g0.x = 1u;                                            // count=1, user descriptor
    g0.y = lds_base;                                      // lds_addr (bytes)
    g0.z = (unsigned int)(ga & 0xFFFFFFFFull);            // global_addr[31:0]
    g0.w = (unsigned int)((ga >> 32) & 0x01FFFFFFull)     // global_addr[56:32]
         | (2u << 30);                                    // type = 2 ("image")
    g1[0] = (2 << 16);                 // data_size = 2 (4 bytes); mask/flags = 0
    g1[1] = (n & 0xFFFF) << 16;        // tensor_dim0[15:0]  @ bits 63:48
    g1[2] = (n >> 16) | (1 << 16);     // tensor_dim0[31:16] @79:64, tensor_dim1=1 @95:80
    g1[3] = (n & 0xFFFF) << 16;        // tile_dim0 = n      @ bits 127:112
    g1[4] = 1;                         // tile_dim1 = 1      @ bits 143:128
    g1[5] = n;                         // tensor_dim0_stride[31:0]
    g1[6] = 0;
    g1[7] = 0;
}
#endif

// ---------------- per-segment sparsemax: one workgroup per segment -----------
__global__ __launch_bounds__(TPB) void sparsemax_seg_kernel(
        const float* __restrict__ x,
        const long long* __restrict__ sizes,
        const unsigned long long* __restrict__ offs,
        float* __restrict__ out) {
    __shared__ float smem[MAXN];
    __shared__ float red[NWAVES + 1];

    const int  b   = blockIdx.x;
    const int  t   = threadIdx.x;
    const int  n   = (int)sizes[b];                 // 2048..8192, multiple of TPB
    const unsigned long long off = offs[b];

    // ---- segment copy global -> LDS via Tensor Data Mover (wave 0 issues) ----
#if HAVE_TDM
    if (t < 32) {
        u32x4 g0; i32x8 g1;
        make_tdm_desc(g0, g1, x + off, (unsigned)(uintptr_t)(&smem[0]), n);
        i32x4 z4 = {0, 0, 0, 0};
#if __clang_major__ >= 23
        i32x8 z8 = {0, 0, 0, 0, 0, 0, 0, 0};
        __builtin_amdgcn_tensor_load_to_lds(g0, g1, z4, z4, z8, 0);
#else
        __builtin_amdgcn_tensor_load_to_lds(g0, g1, z4, z4, 0);
#endif
        __builtin_amdgcn_s_wait_tensorcnt(0);
    }
    __syncthreads();
#else
    for (int i = t; i < n; i += TPB) smem[i] = x[off + i];
    __syncthreads();
#endif

    // ---- pull this thread's slice into registers (n is a multiple of TPB) ----
    const int cnt = n / TPB;                // 4, 8, 12 or 16
    float v[MAXN / TPB];
    float m = -3.402823466e38f;
    #pragma unroll
    for (int i = 0; i < MAXN / TPB; ++i) {
        if (i < cnt) {
            v[i] = smem[t + i * TPB];
            m = fmaxf(m, v[i]);
        }
    }
    m = block_max(m, red);

    // ---- bisection on tau in [max-1, max]: f(tau)=sum(max(x-tau,0)) is dec. --
    float lo = m - 1.0f, hi = m;
    for (int it = 0; it < 24; ++it) {
        float mid = 0.5f * (lo + hi);
        float s = 0.0f;
        #pragma unroll
        for (int i = 0; i < MAXN / TPB; ++i)
            if (i < cnt) s += fmaxf(v[i] - mid, 0.0f);
        s = block_sum(s, red);
        if (s >= 1.0f) lo = mid; else hi = mid;   // uniform branch
    }
    float tau = lo;

    // ---- exact refinement: tau = (sum_{x>tau} x - 1) / |{x>tau}| (x2) -------
    for (int r = 0; r < 2; ++r) {
        float s = 0.0f, k = 0.0f;
        #pragma unroll
        for (int i = 0; i < MAXN / TPB; ++i)
            if (i < cnt && v[i] > tau) { s += v[i]; k += 1.0f; }
        s = block_sum(s, red);
        k = block_sum(k, red);
        if (k > 0.0f) tau = (s - 1.0f) / k;
    }

    // ---- p = n * max(x - tau, 0): write tile into LDS, DMA it back to HBM ---
    const float scale = (float)n;
#if HAVE_TDM
    #pragma unroll
    for (int i = 0; i < MAXN / TPB; ++i)
        if (i < cnt) smem[t + i * TPB] = fmaxf(v[i] - tau, 0.0f) * scale;
    __syncthreads();                         // LDS tile complete before DMA reads it
    if (t < 32) {
        u32x4 g0; i32x8 g1;
        make_tdm_desc(g0, g1, out + off, (unsigned)(uintptr_t)(&smem[0]), n);
        i32x4 z4 = {0, 0, 0, 0};
#if __clang_major__ >= 23
        i32x8 z8 = {0, 0, 0, 0, 0, 0, 0, 0};
        __builtin_amdgcn_tensor_store_from_lds(g0, g1, z4, z4, z8, 0);
#else
        __builtin_amdgcn_tensor_store_from_lds(g0, g1, z4, z4, 0);
#endif
        // no explicit wait: implicit S_WAIT_IDLE at S_ENDPGM drains TENSORcnt,
        // so the DMA overlaps with the tail of this workgroup / next dispatch wavefronts
    }
#else
    #pragma unroll
    for (int i = 0; i < MAXN / TPB; ++i)
        if (i < cnt) out[off + t + i * TPB] = fmaxf(v[i] - tau, 0.0f) * scale;
#endif
}

extern "C" void kernel_launch(void* const* d_in, const int* in_sizes, int n_in,
                              void* d_out, int out_size, void* d_ws, size_t ws_size,
                              hipStream_t stream) {
    const float*     x     = (const float*)d_in[0];
    const long long* sizes = (const long long*)d_in[1];   // int64 graph_size_list
    float*           out   = (float*)d_out;
    const int        B     = in_sizes[1];                 // 2048 segments

    unsigned long long* offs = (unsigned long long*)d_ws; // B * 8 bytes scratch

    seg_offsets_kernel<<<1, B / 2, 0, stream>>>(sizes, offs);
    sparsemax_seg_kernel<<<B, TPB, 0, stream>>>(x, sizes, offs, out);
}